// Voxelizer_62096637165637
// MI455X (gfx1250) — compile-verified
//
#include <hip/hip_runtime.h>
#include <stdint.h>

// Problem constants (match reference exactly)
#define B_      4
#define N_      200000
#define GX_     400
#define GYD_    400
#define NCELL   (GX_ * GYD_)      // 160000 cells per batch (GZ == 1)
#define MAXV    40000
#define MAXP    32
#define INF_IDX 0xFFFFFFFFu

// ---- helpers -------------------------------------------------------------

// Compute validity and cell hash with f32 math identical to the reference:
//   valid = mask && all(PC_MIN <= xyz < PC_MAX)
//   vidx  = clip(int((xyz - PC_MIN)/VOXEL_SIZE), 0, GRID-1)
//   h     = vx*(GY*GZ) + vy*GZ + vz   (GY*GZ = 400, GZ = 1, vz always 0)
__device__ __forceinline__ bool point_cell(const float4 p, bool m, int& h)
{
    bool valid = m
        && (p.x >= -50.0f) && (p.x < 50.0f)
        && (p.y >= -50.0f) && (p.y < 50.0f)
        && (p.z >=  -5.0f) && (p.z <  3.0f);
    if (!valid) return false;
    int vx = (int)((p.x - (-50.0f)) / 0.25f);
    int vy = (int)((p.y - (-50.0f)) / 0.25f);
    vx = min(max(vx, 0), GX_  - 1);
    vy = min(max(vy, 0), GYD_ - 1);
    h = vx * GYD_ + vy;   // vz term is 0
    return true;
}

// ---- kernel 1: bin points, record per-point cell id, count per cell ------
// Stores hEnc[gid] = cell hash for valid points, -1 otherwise, so the later
// insert pass never re-reads points/mask or redoes the f32 binning.

__global__ void k_count(const float4* __restrict__ pts,
                        const unsigned char* __restrict__ mask,
                        int* __restrict__ cellCount,
                        int* __restrict__ hEnc)
{
    int gid = blockIdx.x * blockDim.x + threadIdx.x;
    if (gid >= B_ * N_) return;
    int b = gid / N_;
    float4 p = pts[gid];                       // coalesced global_load_b96/128
    int h;
    if (point_cell(p, mask[gid] != 0, h)) {
        hEnc[gid] = h;
        atomicAdd(&cellCount[b * NCELL + h], 1);
    } else {
        hEnc[gid] = -1;
    }
}

// ---- kernel 2: per-batch exclusive scan over occupied cells --------------
// One 1024-thread workgroup per batch. Each thread owns a contiguous range
// of 157 cells, counts occupied, LDS Hillis-Steele scan over 1024 totals,
// then walks its range assigning voxel ids in ascending cell order.
// Also emits coords and num_points for kept voxels (vid < 40000).

__global__ void k_scan(const int* __restrict__ cellCount,
                       int* __restrict__ vidMap,
                       int* __restrict__ coordsOut,  // [B][MAXV][3] int32
                       int* __restrict__ npOut)      // [B][MAXV]    int32
{
    __shared__ int sdata[1024];
    const int b = blockIdx.x;
    const int t = threadIdx.x;
    const int R = (NCELL + 1023) / 1024;       // 157
    const int lo = t * R;
    const int hi = min(lo + R, NCELL);

    const int* cc = cellCount + b * NCELL;
    int cnt = 0;
    for (int c = lo; c < hi; ++c) cnt += (cc[c] > 0);
    sdata[t] = cnt;
    __syncthreads();

    // inclusive scan (10 steps)
    for (int off = 1; off < 1024; off <<= 1) {
        int u = (t >= off) ? sdata[t - off] : 0;
        __syncthreads();
        sdata[t] += u;
        __syncthreads();
    }

    int vid = sdata[t] - cnt;                  // exclusive prefix
    int* vm = vidMap + b * NCELL;
    for (int c = lo; c < hi; ++c) {
        int n = cc[c];
        if (n > 0) {
            vm[c] = vid;
            if (vid < MAXV) {
                int base = b * MAXV + vid;
                coordsOut[base * 3 + 0] = c / GYD_;   // vx
                coordsOut[base * 3 + 1] = c % GYD_;   // vy
                coordsOut[base * 3 + 2] = 0;          // vz
                npOut[base] = min(n, MAXP);
            }
            ++vid;
        } else {
            vm[c] = -1;
        }
    }
}

// ---- kernel 3: lock-free sorted top-32 insertion via atomicMin cascade ---
// Each valid point inserts its (unique) index into its voxel's 32-slot list.
// Displaced values are carried rightward; at quiescence the slots hold the
// 32 smallest indices in ascending order == the reference's stable ranks.
// Final state is order-independent => deterministic.
// Reads only hEnc + vidMap (no point/mask re-read, no re-binning).

__global__ void k_insert(const int* __restrict__ hEnc,
                         const int* __restrict__ vidMap,
                         unsigned int* __restrict__ slots)
{
    int gid = blockIdx.x * blockDim.x + threadIdx.x;
    if (gid >= B_ * N_) return;
    int h = hEnc[gid];                         // coalesced b32
    if (h < 0) return;
    int b = gid / N_;
    int i = gid - b * N_;
    int vid = vidMap[b * NCELL + h];
    if ((unsigned)vid >= (unsigned)MAXV) return;   // overflow voxels

    unsigned carry = (unsigned)i;
    unsigned* s = slots + ((size_t)(b * MAXV + vid)) * MAXP;
    #pragma unroll 1
    for (int k = 0; k < MAXP; ++k) {
        if (carry == INF_IDX) break;
        unsigned old = atomicMin(&s[k], carry);
        if (old > carry) carry = old;              // displaced value moves on
    }
}

// ---- kernel 4: gather points into the dense voxels tensor ----------------
// Each thread handles 4 consecutive slots: one b128 slot load, then 4
// independent point gathers (MLP) and 4 coalesced b128 stores of the 82MB
// output tensor.

__global__ void k_gather(const uint4* __restrict__ slots4,
                         const float4* __restrict__ pts,
                         float4* __restrict__ voxOut)
{
    const int nQuads = B_ * MAXV * MAXP / 4;   // 1,280,000
    int tid = blockIdx.x * blockDim.x + threadIdx.x;
    if (tid >= nQuads) return;
    int b = tid / (MAXV * MAXP / 4);
    uint4 s = slots4[tid];                     // b128 slot load

    const float4* pb = pts + (size_t)b * N_;
    float4 z = make_float4(0.f, 0.f, 0.f, 0.f);
    float4 v0 = (s.x != INF_IDX) ? pb[(int)s.x] : z;
    float4 v1 = (s.y != INF_IDX) ? pb[(int)s.y] : z;
    float4 v2 = (s.z != INF_IDX) ? pb[(int)s.z] : z;
    float4 v3 = (s.w != INF_IDX) ? pb[(int)s.w] : z;

    float4* o = voxOut + (size_t)tid * 4;
    o[0] = v0; o[1] = v1; o[2] = v2; o[3] = v3;
}

// ---- launch --------------------------------------------------------------

extern "C" void kernel_launch(void* const* d_in, const int* in_sizes, int n_in,
                              void* d_out, int out_size, void* d_ws, size_t ws_size,
                              hipStream_t stream)
{
    (void)in_sizes; (void)n_in; (void)out_size; (void)ws_size;

    const float4*        pts  = (const float4*)d_in[0];        // (B,N,4) f32
    const unsigned char* mask = (const unsigned char*)d_in[1]; // (B,N) bool

    // Output layout (flat, return order): voxels f32 | coords i32 | num_points i32
    float*  out       = (float*)d_out;
    float4* voxOut    = (float4*)out;                                   // B*MAXV*MAXP float4
    size_t  voxElems  = (size_t)B_ * MAXV * MAXP * 4;                   // 20,480,000
    int*    coordsOut = (int*)(out + voxElems);                         // B*MAXV*3
    int*    npOut     = (int*)(out + voxElems + (size_t)B_ * MAXV * 3); // B*MAXV

    // Workspace layout
    char*         ws        = (char*)d_ws;
    size_t        off       = 0;
    int*          cellCount = (int*)(ws + off); off += (size_t)B_ * NCELL * 4;      // 2.56 MB
    int*          vidMap    = (int*)(ws + off); off += (size_t)B_ * NCELL * 4;      // 2.56 MB
    int*          hEnc      = (int*)(ws + off); off += (size_t)B_ * N_ * 4;         // 3.20 MB
    unsigned int* slots     = (unsigned int*)(ws + off);                            // 20.48 MB

    // Stream-ordered inits (graph-capture legal)
    hipMemsetAsync(cellCount, 0,    (size_t)B_ * NCELL * 4, stream);
    hipMemsetAsync(slots,     0xFF, (size_t)B_ * MAXV * MAXP * 4, stream);
    // coords (B*MAXV*3) and num_points (B*MAXV) are contiguous: 16 B / voxel
    hipMemsetAsync(coordsOut, 0,    (size_t)B_ * MAXV * 16, stream);

    const int totalPts = B_ * N_;
    k_count <<<(totalPts + 255) / 256, 256, 0, stream>>>(pts, mask, cellCount, hEnc);
    k_scan  <<<B_, 1024, 0, stream>>>(cellCount, vidMap, coordsOut, npOut);
    k_insert<<<(totalPts + 255) / 256, 256, 0, stream>>>(hEnc, vidMap, slots);

    const int nQuads = B_ * MAXV * MAXP / 4;
    k_gather<<<(nQuads + 255) / 256, 256, 0, stream>>>((const uint4*)slots, pts, voxOut);
}